// DAHNRec_Encoder_90941637525622
// MI455X (gfx1250) — compile-verified
//
#include <hip/hip_runtime.h>
#include <hip/hip_bf16.h>

#define U_CNT   100000
#define I_CNT   50000
#define N_CNT   150000
#define NNZ_CNT 3200000
#define EMB     64
#define N_LAYERS 3
#define EPS_NOISE 0.1f
#define BN_EPS    1e-5f
#define LRELU     0.01f
#define EDGES_PER_WAVE 256

typedef __attribute__((ext_vector_type(2))) float v2f;
typedef __attribute__((ext_vector_type(8))) float v8f;

// ---------------------------------------------------------------------------
// SpMM: out[r,:] += vals[e] * x[cols[e],:]   (COO, rows sorted ascending)
// One wave handles EDGES_PER_WAVE consecutive edges; lane l owns columns
// {2l, 2l+1}. Accumulate locally while the row is unchanged (avg run ~21),
// flush with global f32 atomics on row change / span end.
// ---------------------------------------------------------------------------
__global__ void spmm_kernel(const int* __restrict__ rows,
                            const int* __restrict__ cols,
                            const float* __restrict__ vals,
                            const float* __restrict__ x,
                            float* __restrict__ out, int nnz) {
    const long long wave = (long long)((blockIdx.x * blockDim.x + threadIdx.x) >> 5);
    const int lane = threadIdx.x & 31;
    long long e0 = wave * EDGES_PER_WAVE;
    if (e0 >= nnz) return;
    long long e1 = e0 + EDGES_PER_WAVE;
    if (e1 > nnz) e1 = nnz;

    const int c = lane * 2;
    float a0 = 0.f, a1 = 0.f;
    int cur_row = rows[e0];
    for (long long e = e0; e < e1; ++e) {
        const int r = rows[e];
        if (r != cur_row) {
            atomicAdd(&out[(size_t)cur_row * EMB + c],     a0);
            atomicAdd(&out[(size_t)cur_row * EMB + c + 1], a1);
            a0 = 0.f; a1 = 0.f; cur_row = r;
        }
        const int cl = cols[e];
        const float v = vals[e];
        const float* xr = x + (size_t)cl * EMB + c;
        a0 = fmaf(v, xr[0], a0);
        a1 = fmaf(v, xr[1], a1);
    }
    atomicAdd(&out[(size_t)cur_row * EMB + c],     a0);
    atomicAdd(&out[(size_t)cur_row * EMB + c + 1], a1);
}

// ---------------------------------------------------------------------------
// h = LeakyReLU(X @ W^T + b), W selected per partition (row tiles of 16 never
// cross the U boundary since U % 16 == 0). One wave = one 16x16 output tile,
// K=64 via 16 chained V_WMMA_F32_16X16X4_F32. Block = 128 threads = 4 waves
// covering the 4 column tiles of one 16-row stripe.
//
// A (16x4 f32) lane layout: lanes 0-15 -> K={k0,k0+1}, lanes 16-31 -> {k0+2,k0+3}
// B (4x16 f32): same K split, lane%16 selects output column n.
// C/D (16x16 f32): VGPR v, lane<16 -> (M=v, N=lane); lane>=16 -> (M=v+8, N=lane-16)
// ---------------------------------------------------------------------------
__global__ void linear_lrelu_wmma(const float* __restrict__ X,
                                  const float* __restrict__ Wu, const float* __restrict__ bu,
                                  const float* __restrict__ Wi, const float* __restrict__ bi,
                                  float* __restrict__ H) {
    const int lane   = threadIdx.x & 31;
    const int waveId = threadIdx.x >> 5;          // 0..3 -> column tile
    const int r0 = blockIdx.x * 16;
    const int n0 = waveId * 16;
    const bool isUser = (r0 < U_CNT);
    const float* __restrict__ W = isUser ? Wu : Wi;
    const float* __restrict__ B = isUser ? bu : bi;

    const int lhalf = lane & 15;
    const int khi   = (lane >> 4) << 1;           // 0 or 2

    const float* xr = X + (size_t)(r0 + lhalf) * EMB + khi;
    const float* wr = W + (size_t)(n0 + lhalf) * EMB + khi;   // W[n][k] -> B[k][n]

    v8f cacc = {};
#pragma unroll
    for (int k0 = 0; k0 < EMB; k0 += 4) {
        v2f a, b;
        a.x = xr[k0]; a.y = xr[k0 + 1];
        b.x = wr[k0]; b.y = wr[k0 + 1];
        cacc = __builtin_amdgcn_wmma_f32_16x16x4_f32(
            /*neg_a=*/false, a, /*neg_b=*/false, b,
            /*c_mod=*/(short)0, cacc, /*reuse_a=*/false, /*reuse_b=*/false);
    }

    const int col  = n0 + lhalf;
    const float bv = B[col];
    const int rbase = r0 + ((lane >> 4) << 3);    // +0 or +8
#pragma unroll
    for (int v = 0; v < 8; ++v) {
        float val = cacc[v] + bv;
        val = (val >= 0.f) ? val : LRELU * val;
        H[(size_t)(rbase + v) * EMB + col] = val;
    }
}

// ---------------------------------------------------------------------------
// Column sum / sum-of-squares over a row range (users or items).
// Block = 256 threads = 4 row-groups x 64 columns; LDS tree reduce, then one
// atomicAdd per column per block.
// ---------------------------------------------------------------------------
__global__ void stats_kernel(const float* __restrict__ h, float* __restrict__ stats,
                             int rowStart, int rowEnd, int statOff) {
    __shared__ float s1[256];
    __shared__ float s2[256];
    const int tid  = threadIdx.x;
    const int col  = tid & 63;
    const int rgrp = tid >> 6;  // 0..3
    float sum = 0.f, sumsq = 0.f;
    for (int r = rowStart + blockIdx.x * 4 + rgrp; r < rowEnd; r += gridDim.x * 4) {
        const float v = h[(size_t)r * EMB + col];
        sum += v;
        sumsq = fmaf(v, v, sumsq);
    }
    s1[tid] = sum; s2[tid] = sumsq;
    __syncthreads();
    if (tid < 128) { s1[tid] += s1[tid + 128]; s2[tid] += s2[tid + 128]; }
    __syncthreads();
    if (tid < 64) {
        atomicAdd(&stats[statOff + tid],      s1[tid] + s1[tid + 64]);
        atomicAdd(&stats[statOff + 64 + tid], s2[tid] + s2[tid + 64]);
    }
}

// ---------------------------------------------------------------------------
// Fold BN stats + gamma/beta into per-column scale/shift:
//   scale = g * rsqrt(var + eps); shift = bt - mean * scale
// ss layout: [0:64] scale_u [64:128] shift_u [128:192] scale_i [192:256] shift_i
// ---------------------------------------------------------------------------
__global__ void bn_finalize(const float* __restrict__ stats,
                            const float* __restrict__ gu, const float* __restrict__ btu,
                            const float* __restrict__ gi, const float* __restrict__ bti,
                            float* __restrict__ ss) {
    const int tid = threadIdx.x;
    if (tid >= 128) return;
    const int c = tid & 63;
    const bool usr = tid < 64;
    const float* base = stats + (usr ? 0 : 128);
    const float cnt = usr ? (float)U_CNT : (float)I_CNT;
    const float mean = base[c] / cnt;
    float var = base[64 + c] / cnt - mean * mean;
    const float rstd = rsqrtf(var + BN_EPS);
    const float g  = (usr ? gu : gi)[c];
    const float bt = (usr ? btu : bti)[c];
    const float sc = g * rstd;
    ss[(usr ? 0 : 128) + c]      = sc;
    ss[(usr ? 64 : 192) + c]     = bt - mean * sc;
}

// ---------------------------------------------------------------------------
// Per-row: affine-BN, L2 normalize (wave32 shuffle reduction over 64 elems,
// 2 per lane), ego = s + sign(s)*noise*EPS; acc init/accumulate; emb_cl on
// first layer. One wave per row.
// ---------------------------------------------------------------------------
__global__ void apply_kernel(const float* __restrict__ sp, const float* __restrict__ h,
                             const float* __restrict__ ss, float* __restrict__ ego,
                             float* __restrict__ acc, float* __restrict__ embcl, int first) {
    const int wave = (blockIdx.x * blockDim.x + threadIdx.x) >> 5;
    const int lane = threadIdx.x & 31;
    if (wave >= N_CNT) return;
    const size_t base = (size_t)wave * EMB + lane * 2;
    const int part = (wave < U_CNT) ? 0 : 128;

    const float sc0 = ss[part + lane * 2],      sc1 = ss[part + lane * 2 + 1];
    const float sh0 = ss[part + 64 + lane * 2], sh1 = ss[part + 64 + lane * 2 + 1];

    const float n0 = fmaf(h[base],     sc0, sh0);
    const float n1 = fmaf(h[base + 1], sc1, sh1);

    float p = n0 * n0 + n1 * n1;
#pragma unroll
    for (int off = 16; off >= 1; off >>= 1) p += __shfl_xor(p, off, 32);
    const float inv = EPS_NOISE / fmaxf(sqrtf(p), 1e-12f);

    const float s0 = sp[base], s1 = sp[base + 1];
    const float sg0 = (s0 > 0.f) ? 1.f : ((s0 < 0.f) ? -1.f : 0.f);
    const float sg1 = (s1 > 0.f) ? 1.f : ((s1 < 0.f) ? -1.f : 0.f);
    const float e0 = fmaf(sg0 * n0, inv, s0);
    const float e1 = fmaf(sg1 * n1, inv, s1);

    ego[base] = e0; ego[base + 1] = e1;
    const float a0 = first ? e0 : acc[base] + e0;
    const float a1 = first ? e1 : acc[base + 1] + e1;
    acc[base] = a0; acc[base + 1] = a1;
    if (first) { embcl[base] = e0; embcl[base + 1] = e1; }
}

__global__ void final_kernel(const float* __restrict__ acc, float* __restrict__ out, size_t n) {
    const size_t i = (size_t)blockIdx.x * blockDim.x + threadIdx.x;
    if (i < n) out[i] = acc[i] * (1.0f / 3.0f);
}

// ---------------------------------------------------------------------------
extern "C" void kernel_launch(void* const* d_in, const int* in_sizes, int n_in,
                              void* d_out, int out_size, void* d_ws, size_t ws_size,
                              hipStream_t stream) {
    const float* user_emb = (const float*)d_in[0];
    const float* item_emb = (const float*)d_in[1];
    const int*   rows     = (const int*)d_in[2];
    const int*   cols     = (const int*)d_in[3];
    const float* vals     = (const float*)d_in[4];
    const float* Wu  = (const float*)d_in[5];
    const float* bu  = (const float*)d_in[6];
    const float* gu  = (const float*)d_in[7];
    const float* btu = (const float*)d_in[8];
    const float* Wi  = (const float*)d_in[9];
    const float* bi  = (const float*)d_in[10];
    const float* gi  = (const float*)d_in[11];
    const float* bti = (const float*)d_in[12];
    float* out = (float*)d_out;

    const size_t NE = (size_t)N_CNT * EMB;
    float* ego   = (float*)d_ws;
    float* sp    = ego + NE;
    float* h     = sp + NE;
    float* acc   = h + NE;
    float* stats = acc + NE;      // 256 floats: sum_u | sumsq_u | sum_i | sumsq_i
    float* ss    = stats + 256;   // 256 floats: scale/shift per partition

    // ego = concat(user_emb, item_emb)
    hipMemcpyAsync(ego, user_emb, (size_t)U_CNT * EMB * sizeof(float),
                   hipMemcpyDeviceToDevice, stream);
    hipMemcpyAsync(ego + (size_t)U_CNT * EMB, item_emb, (size_t)I_CNT * EMB * sizeof(float),
                   hipMemcpyDeviceToDevice, stream);

    const int spmmWaves   = (NNZ_CNT + EDGES_PER_WAVE - 1) / EDGES_PER_WAVE;
    const int spmmBlocks  = (spmmWaves * 32 + 255) / 256;
    const int linBlocks   = N_CNT / 16;                 // 9375, each 4 waves
    const int applyBlocks = (N_CNT * 32 + 255) / 256;   // one wave per row

    for (int k = 0; k < N_LAYERS; ++k) {
        hipMemsetAsync(sp, 0, NE * sizeof(float), stream);
        spmm_kernel<<<spmmBlocks, 256, 0, stream>>>(rows, cols, vals, ego, sp, NNZ_CNT);

        hipMemsetAsync(stats, 0, 256 * sizeof(float), stream);
        linear_lrelu_wmma<<<linBlocks, 128, 0, stream>>>(sp, Wu, bu, Wi, bi, h);

        stats_kernel<<<512, 256, 0, stream>>>(h, stats, 0, U_CNT, 0);
        stats_kernel<<<256, 256, 0, stream>>>(h, stats, U_CNT, N_CNT, 128);
        bn_finalize<<<1, 128, 0, stream>>>(stats, gu, btu, gi, bti, ss);

        apply_kernel<<<applyBlocks, 256, 0, stream>>>(sp, h, ss, ego, acc,
                                                      out + NE, (k == 0) ? 1 : 0);
    }
    final_kernel<<<(int)((NE + 255) / 256), 256, 0, stream>>>(acc, out, NE);
}